// ROIAlign_84808424227330
// MI455X (gfx1250) — compile-verified
//
#include <hip/hip_runtime.h>
#include <stdint.h>

// ---------------------------------------------------------------------------
// ROI Align (bilinear, 2x2 sample grid -> max pool) for MI455X / gfx1250.
//
// Memory-bound op: 51.4 MB stores vs 5.1 MB L2-resident features => ~2.2 us
// HBM bound. No matmul structure (per-sample gather weights), so WMMA is the
// wrong tool. We instead use the CDNA5 Tensor Data Mover to asynchronously
// stage each ROI's <=18x18 feature patch (per 16-channel chunk, strided 3D
// tile) into LDS, double-buffered with s_wait_tensorcnt, and compute the
// bilinear + maxpool from LDS.
// ---------------------------------------------------------------------------

#define C_DIM   512
#define H_DIM   50
#define W_DIM   50
#define PH      7
#define PW      7
#define HS      14          // PH * SS
#define WS      14          // PW * SS
#define NSAMP   (HS * WS)   // 196
#define CH_CHUNK 16
#define N_CHUNK (C_DIM / CH_CHUNK)   // 32
#define MAX_PDIM 18
#define MAX_PATCH (MAX_PDIM * MAX_PDIM)  // 324 floats per channel
#define BLOCK   256

typedef __attribute__((ext_vector_type(4))) unsigned int v4u;
typedef __attribute__((ext_vector_type(8))) int          v8i;
typedef __attribute__((ext_vector_type(4))) int          v4i;

__device__ __forceinline__ unsigned rfl(unsigned x) {
    return (unsigned)__builtin_amdgcn_readfirstlane((int)x);
}

// Issue one TDM 3D-tile load: global feature patch -> LDS.
// Descriptor packing per CDNA5 ISA.
__device__ __forceinline__ void tdm_load_patch(const float* feat,
                                               unsigned lds_byte_addr,
                                               int c0, int py0, int px0,
                                               int patch_w, int patch_h)
{
    unsigned long long ga = (unsigned long long)(uintptr_t)feat
        + 4ull * (unsigned long long)(c0 * (H_DIM * W_DIM) + py0 * W_DIM + px0);
    // Remaining-extent tensor dims from tile start (tile always in-bounds,
    // so OOB-zeroing never triggers).
    unsigned td0 = (unsigned)(W_DIM - px0);
    unsigned td1 = (unsigned)(H_DIM - py0);

    // D# group 0: count=1 | lds_addr | global_addr[56:0] | type=2
    v4u g0;
    g0.x = rfl(1u);
    g0.y = rfl(lds_byte_addr);
    g0.z = rfl((unsigned)(ga & 0xFFFFFFFFull));
    g0.w = rfl((unsigned)((ga >> 32) & 0x1FFFFFFull) | (2u << 30));

    // D# group 1: data_size=4B, tensor_dim0/1, tile_dim0/1/2, strides 50/2500
    v8i g1;
    g1[0] = (int)rfl(2u << 16);                                        // data_size=2 (4B)
    g1[1] = (int)rfl((td0 & 0xFFFFu) << 16);                           // tdim0 lo
    g1[2] = (int)rfl(((td0 >> 16) & 0xFFFFu) | ((td1 & 0xFFFFu) << 16));
    g1[3] = (int)rfl(((td1 >> 16) & 0xFFFFu) | ((unsigned)patch_w << 16)); // tile_dim0
    g1[4] = (int)rfl((unsigned)patch_h | ((unsigned)CH_CHUNK << 16));  // tile_dim1|2
    g1[5] = (int)rfl((unsigned)W_DIM);                                 // dim0 stride lo
    g1[6] = (int)rfl(((unsigned)(H_DIM * W_DIM) & 0xFFFFu) << 16);     // dim1 stride lo
    g1[7] = (int)rfl((unsigned)(H_DIM * W_DIM) >> 16);                 // dim1 stride hi

    // D# group 2: tensor_dim2 = CH_CHUNK, tensor_dim2_stride = H*W
    v4i g2;
    g2.x = (int)rfl((unsigned)CH_CHUNK);
    g2.y = 0;
    g2.z = (int)rfl((unsigned)(H_DIM * W_DIM));
    g2.w = 0;

    v4i g3 = {0, 0, 0, 0};

    // 6-arg toolchain variant: extra int32x8 descriptor slot (unused -> 0).
    v8i gx = {0, 0, 0, 0, 0, 0, 0, 0};

    __builtin_amdgcn_tensor_load_to_lds(g0, g1, g2, g3, gx, /*cpol=*/0);
}

__global__ __launch_bounds__(BLOCK)
void roialign_tdm_kernel(const float* __restrict__ feat,
                         const float* __restrict__ rois,
                         const float* __restrict__ ratio_p,
                         float* __restrict__ out)
{
    __shared__ float s_patch[2][CH_CHUNK * MAX_PATCH];  // 41472 B
    __shared__ int   s_off[NSAMP][4];                   // 3136 B
    __shared__ float s_wgt[NSAMP][4];                   // 3136 B

    const int n   = blockIdx.x;
    const int tid = threadIdx.x;
    const float ratio = ratio_p[0];

    // ---- uniform per-ROI geometry -------------------------------------
    const float y0f = rois[n * 4 + 0] * ratio;
    const float x0f = rois[n * 4 + 1] * ratio;
    const float y1f = rois[n * 4 + 2] * ratio;
    const float x1f = rois[n * 4 + 3] * ratio;
    const float hstep = (y1f - y0f) * (1.0f / HS);
    const float wstep = (x1f - x0f) * (1.0f / WS);

    const float ymin = y0f + 0.5f * hstep;
    const float ymax = y0f + (HS - 0.5f) * hstep;
    const float xmin = x0f + 0.5f * wstep;
    const float xmax = x0f + (WS - 0.5f) * wstep;
    const int py0 = min(max((int)floorf(ymin), 0), H_DIM - 1);
    const int py1 = min(max((int)ceilf(ymax),  0), H_DIM - 1);
    const int px0 = min(max((int)floorf(xmin), 0), W_DIM - 1);
    const int px1 = min(max((int)ceilf(xmax),  0), W_DIM - 1);
    const int patch_h = min(py1 - py0 + 1, MAX_PDIM);
    const int patch_w = min(px1 - px0 + 1, MAX_PDIM);
    const int patch_sz = patch_h * patch_w;

    // ---- phase 1: per-sample tap offsets + bilinear weights -----------
    if (tid < NSAMP) {
        const int sy = tid / WS;
        const int sx = tid - sy * WS;
        const float y = y0f + (sy + 0.5f) * hstep;
        const float x = x0f + (sx + 0.5f) * wstep;
        const float fy = floorf(y), fx = floorf(x);
        const float ly = y - fy,    lx = x - fx;
        const int iu  = min(max((int)fy, 0), H_DIM - 1);
        const int idn = min(max((int)ceilf(y), 0), H_DIM - 1);
        const int il  = min(max((int)fx, 0), W_DIM - 1);
        const int ir  = min(max((int)ceilf(x), 0), W_DIM - 1);
        const int ru = iu - py0, rd = idn - py0;
        const int rl = il - px0, rr = ir - px0;
        s_off[tid][0] = ru * patch_w + rl;
        s_off[tid][1] = ru * patch_w + rr;
        s_off[tid][2] = rd * patch_w + rl;
        s_off[tid][3] = rd * patch_w + rr;
        s_wgt[tid][0] = (1.0f - ly) * (1.0f - lx);
        s_wgt[tid][1] = (1.0f - ly) * lx;
        s_wgt[tid][2] = ly * (1.0f - lx);
        s_wgt[tid][3] = ly * lx;
    }

    const unsigned lds0 = (unsigned)(uintptr_t)&s_patch[0][0];
    const unsigned lds1 = (unsigned)(uintptr_t)&s_patch[1][0];

    // Prime the pipeline: TDM load of chunk 0 (wave 0 only; TDM ignores EXEC,
    // so gate with a whole-wave branch).
    if (tid < 32) {
        tdm_load_patch(feat, lds0, 0, py0, px0, patch_w, patch_h);
    }
    __syncthreads();   // also publishes s_off / s_wgt

    // ---- phase 2: double-buffered TDM + compute -----------------------
    for (int chunk = 0; chunk < N_CHUNK; ++chunk) {
        if (tid < 32) {
            if (chunk + 1 < N_CHUNK) {
                tdm_load_patch(feat, ((chunk + 1) & 1) ? lds1 : lds0,
                               (chunk + 1) * CH_CHUNK, py0, px0,
                               patch_w, patch_h);
                __builtin_amdgcn_s_wait_tensorcnt(1);  // chunk's data landed
            } else {
                __builtin_amdgcn_s_wait_tensorcnt(0);
            }
        }
        __syncthreads();   // publish chunk's patch to all 8 waves

        const float* P = (chunk & 1) ? &s_patch[1][0] : &s_patch[0][0];

        for (int i = tid; i < CH_CHUNK * PH * PW; i += BLOCK) {
            const int cl = i / (PH * PW);
            const int q  = i - cl * (PH * PW);
            const int ph = q / PW;
            const int pw = q - ph * PW;
            const float* Pc = P + cl * patch_sz;

            float m = -3.402823466e+38f;
#pragma unroll
            for (int dy = 0; dy < 2; ++dy) {
#pragma unroll
                for (int dx = 0; dx < 2; ++dx) {
                    const int s = (2 * ph + dy) * WS + (2 * pw + dx);
                    const float v = s_wgt[s][0] * Pc[s_off[s][0]]
                                  + s_wgt[s][1] * Pc[s_off[s][1]]
                                  + s_wgt[s][2] * Pc[s_off[s][2]]
                                  + s_wgt[s][3] * Pc[s_off[s][3]];
                    m = fmaxf(m, v);
                }
            }
            out[((size_t)n * C_DIM + (size_t)(chunk * CH_CHUNK + cl)) * (PH * PW) + q] = m;
        }
        __syncthreads();   // buffer may be overwritten two chunks later
    }
}

extern "C" void kernel_launch(void* const* d_in, const int* in_sizes, int n_in,
                              void* d_out, int out_size, void* d_ws, size_t ws_size,
                              hipStream_t stream) {
    const float* feat  = (const float*)d_in[0];
    const float* rois  = (const float*)d_in[1];
    const float* ratio = (const float*)d_in[2];
    float* out = (float*)d_out;
    const int n_rois = in_sizes[1] / 4;   // rois is (N, 4)
    roialign_tdm_kernel<<<n_rois, BLOCK, 0, stream>>>(feat, rois, ratio, out);
}